// ScaledMaxPlus_70832600646266
// MI455X (gfx1250) — compile-verified
//
#include <hip/hip_runtime.h>

// Tropical (max-plus) matmul: out[b,o] = max_i ( W[o,i] + X[b,i] * factors[o] )
// B=512, OUT=1024, IN=1024, fp32.
//
// CDNA5 path: async global->LDS copies (transposing scatter) + s_wait_asynccnt
// double buffering; VALU fma + max3 inner loop (max-plus is not a WMMA semiring).

namespace {
constexpr int kB   = 512;
constexpr int kOut = 1024;
constexpr int kIn  = 1024;

constexpr int BM = 32;           // batch rows per block
constexpr int BN = 64;           // output rows per block
constexpr int BK = 32;           // k-chunk
constexpr int XS = BM + 2;       // 34: LDS row stride for Xs (bank-conflict-free, 8B-aligned reads)
constexpr int WSTR = BN + 2;     // 66: LDS row stride for Ws
constexpr int NTHREADS = 256;    // 8 waves (wave32)
constexpr int NCHUNK = kIn / BK; // 32
// async instructions per wave per chunk: X: BM*BK/256 = 4, W: BN*BK/256 = 8
constexpr int ASYNC_PER_CHUNK = (BM * BK) / NTHREADS + (BN * BK) / NTHREADS; // 12
} // namespace

__device__ __forceinline__ unsigned lds_byte_off(const void* p) {
  // Low 32 bits of a generic pointer into LDS aperture == workgroup-relative
  // LDS byte offset (flat->LDS mapping truncates to addr[31:0]).
  return (unsigned)(unsigned long long)p;
}

__device__ __forceinline__ void async_copy_b32(unsigned lds_off, const float* gaddr) {
  // Per-lane: copies 4 bytes from global address gaddr to LDS offset lds_off, tracked by ASYNCcnt.
  asm volatile("global_load_async_to_lds_b32 %0, %1, off"
               :: "v"(lds_off), "v"(gaddr)
               : "memory");
}

__global__ __launch_bounds__(NTHREADS, 2)
void ScaledMaxPlus_kernel(const float* __restrict__ X,
                          const float* __restrict__ W,
                          const float* __restrict__ F,
                          float* __restrict__ out) {
  __shared__ float Xs[2][BK][XS];    // k-major, transposed X tile
  __shared__ float Ws[2][BK][WSTR];  // k-major, transposed W tile

  const int tid = threadIdx.x;
  const int bx  = blockIdx.x;        // OUT tile index
  const int by  = blockIdx.y;        // B tile index
  const int tx  = tid & 15;          // 16 groups of 4 output cols
  const int ty  = tid >> 4;          // 16 groups of 2 batch rows
  const int n0  = bx * BN;
  const int b0  = by * BM;

  // Per-thread output-column scale factors (factors is (OUT,1) flat).
  const float4 f4 = *(const float4*)(F + n0 + tx * 4);
  const float fj[4] = {f4.x, f4.y, f4.z, f4.w};

  float acc[2][4];
#pragma unroll
  for (int i = 0; i < 2; ++i)
#pragma unroll
    for (int j = 0; j < 4; ++j)
      acc[i][j] = -3.402823466e38f; // -FLT_MAX

  // Issue one k-chunk's async transposing copy (12 async instrs per wave).
  auto prefetch = [&](int c, int buf) {
    const int k0 = c * BK;
    // X tile: global row-major (b, k) -> LDS Xs[k][b]
#pragma unroll
    for (int it = 0; it < (BM * BK) / NTHREADS; ++it) {
      const int e  = tid + it * NTHREADS;
      const int b  = e >> 5;        // row within tile
      const int kk = e & (BK - 1);  // lane-contiguous k -> coalesced global read
      async_copy_b32(lds_byte_off(&Xs[buf][kk][b]),
                     X + (size_t)(b0 + b) * kIn + k0 + kk);
    }
    // W tile: global row-major (n, k) -> LDS Ws[k][n]
#pragma unroll
    for (int it = 0; it < (BN * BK) / NTHREADS; ++it) {
      const int e  = tid + it * NTHREADS;
      const int n  = e >> 5;
      const int kk = e & (BK - 1);
      async_copy_b32(lds_byte_off(&Ws[buf][kk][n]),
                     W + (size_t)(n0 + n) * kIn + k0 + kk);
    }
  };

  prefetch(0, 0);

  for (int c = 0; c < NCHUNK; ++c) {
    const int buf = c & 1;
    if (c + 1 < NCHUNK) {
      // Next chunk in flight (WAR-safe: trailing barrier of iter c-1 freed buf^1),
      // then wait until only its 12 async ops remain -> chunk c is resident.
      prefetch(c + 1, buf ^ 1);
      asm volatile("s_wait_asynccnt %0" :: "i"(ASYNC_PER_CHUNK) : "memory");
    } else {
      asm volatile("s_wait_asynccnt 0" ::: "memory");
    }
    __syncthreads(); // publish this wave's waited-on LDS writes to all waves

    const float* xp = &Xs[buf][0][ty * 2];
    const float* wp = &Ws[buf][0][tx * 4];

#pragma unroll
    for (int k = 0; k < BK; k += 2) {
      const float2 xa  = *(const float2*)(xp + (k + 0) * XS);
      const float2 xb  = *(const float2*)(xp + (k + 1) * XS);
      const float2 wa0 = *(const float2*)(wp + (k + 0) * WSTR);
      const float2 wa1 = *(const float2*)(wp + (k + 0) * WSTR + 2);
      const float2 wb0 = *(const float2*)(wp + (k + 1) * WSTR);
      const float2 wb1 = *(const float2*)(wp + (k + 1) * WSTR + 2);

      const float xr[2][2] = {{xa.x, xa.y}, {xb.x, xb.y}};
      const float wr[2][4] = {{wa0.x, wa0.y, wa1.x, wa1.y},
                              {wb0.x, wb0.y, wb1.x, wb1.y}};
#pragma unroll
      for (int i = 0; i < 2; ++i)
#pragma unroll
        for (int j = 0; j < 4; ++j) {
          const float t0 = fmaf(xr[0][i], fj[j], wr[0][j]);
          const float t1 = fmaf(xr[1][i], fj[j], wr[1][j]);
          acc[i][j] = fmaxf(acc[i][j], fmaxf(t0, t1)); // -> v_max3
        }
    }
    __syncthreads(); // all waves done reading buf before it is overwritten
  }

#pragma unroll
  for (int i = 0; i < 2; ++i) {
    const float4 v = {acc[i][0], acc[i][1], acc[i][2], acc[i][3]};
    *(float4*)(out + (size_t)(b0 + ty * 2 + i) * kOut + n0 + tx * 4) = v;
  }
}

extern "C" void kernel_launch(void* const* d_in, const int* in_sizes, int n_in,
                              void* d_out, int out_size, void* d_ws, size_t ws_size,
                              hipStream_t stream) {
  const float* X = (const float*)d_in[0];        // (B, IN)
  const float* W = (const float*)d_in[1];        // (OUT, IN)
  const float* F = (const float*)d_in[2];        // (OUT, 1) -> flat OUT
  float* out     = (float*)d_out;                // (B, OUT)

  dim3 grid(kOut / BN, kB / BM);                 // (16, 16) = 256 workgroups
  dim3 block(NTHREADS);                          // 8 wave32 waves
  hipLaunchKernelGGL(ScaledMaxPlus_kernel, grid, block, 0, stream, X, W, F, out);
}